// CapsuleSubLayer_51153060496121
// MI455X (gfx1250) — compile-verified
//
#include <hip/hip_runtime.h>
#include <math.h>

// CapsuleSubLayer for MI455X (gfx1250), fp32, memory-bound (~70 MiB @ 23.3 TB/s).
// GEMM u_hat_last = u_last(16384x64) x W7(64x512) via V_WMMA_F32_16X16X4_F32,
// fused with the routing reductions so u_hat_last is never spilled to HBM.

typedef __attribute__((ext_vector_type(2))) float v2f;
typedef __attribute__((ext_vector_type(8))) float v8f;

#define NUM_IN   8
#define NUM_OUT  8
#define IN_DIM   64
#define OUT_DIM  64
#define JB       16384      // bsz*seq = 32*512
#define NCOL     512        // NUM_OUT*OUT_DIM
#define STRIPS   1024       // JB/16
#define CAP_EPS  1e-9f

union V8 { v8f v; float f[8]; };

// ---------------------------------------------------------------------------
// Kernel A: partial sums of x over rows, per (i, part) -> xpart[i*16+part][d]
// x layout: (num_in=8, bsz=32, seq=512, d=64); per i it is 16384 rows x 64.
// ---------------------------------------------------------------------------
__global__ __launch_bounds__(256) void k_xpart(const float* __restrict__ x,
                                               float* __restrict__ xpart) {
    int part = blockIdx.x;          // 0..15
    int i    = blockIdx.y;          // 0..7
    int t    = threadIdx.x;         // 0..255
    int d    = t & 63;
    int g    = t >> 6;              // 0..3
    const float* xi = x + (size_t)i * JB * IN_DIM;
    int r0 = part * 1024;
    float acc = 0.f;
    for (int r = r0 + g; r < r0 + 1024; r += 4)
        acc += xi[(size_t)r * 64 + d];
    __shared__ float sd[256];
    sd[t] = acc;
    __syncthreads();
    if (g == 0)
        xpart[(i * 16 + part) * 64 + d] = sd[d] + sd[64 + d] + sd[128 + d] + sd[192 + d];
}

// ---------------------------------------------------------------------------
// Kernel C: u_mean -> u_hat_mean[i][j][e] = sum_d W[i,j,d,e]*u_mean[i,d]; B=0.
// ---------------------------------------------------------------------------
__global__ __launch_bounds__(256) void k_uhm(const float* __restrict__ xpart,
                                             const float* __restrict__ W,
                                             float* __restrict__ uhm,
                                             float* __restrict__ Bmat) {
    __shared__ float umean[512];
    int t = threadIdx.x;
    for (int v = t; v < 512; v += 256) {
        int i = v >> 6, d = v & 63;
        float s = 0.f;
        for (int p = 0; p < 16; ++p) s += xpart[(i * 16 + p) * 64 + d];
        umean[v] = s * (1.0f / (float)JB);
    }
    __syncthreads();
    for (int v = t; v < 4096; v += 256) {
        int i = v >> 9, j = (v >> 6) & 7, e = v & 63;
        const float* wp = W + ((size_t)i * 8 + j) * 4096 + e;   // W[i,j,d,e], stride_d=64
        const float* um = umean + i * 64;
        float s = 0.f;
        for (int d = 0; d < 64; ++d) s += um[d] * wp[d * 64];
        uhm[v] = s;
    }
    if (t < 64) Bmat[t] = 0.f;
}

// ---------------------------------------------------------------------------
// Kernel R1: column softmax of B (8x8, over i), keep row i=7 -> cvec[j].
// ---------------------------------------------------------------------------
__global__ void k_softmax(const float* __restrict__ Bmat, float* __restrict__ cvec) {
    int j = threadIdx.x;
    if (j < 8) {
        float m = Bmat[j];
        for (int i = 1; i < 8; ++i) m = fmaxf(m, Bmat[i * 8 + j]);
        float den = 0.f;
        for (int i = 0; i < 8; ++i) den += expf(Bmat[i * 8 + j] - m);
        cvec[j] = expf(Bmat[56 + j] - m) / den;
    }
}

// ---------------------------------------------------------------------------
// Fused pass: per block, compute a 16x512 strip of u_hat_last via WMMA into
// LDS, compute squash factors, then either (mode 0) emit per-block vsum
// partials, or (mode 1) write the permuted final output.
// 8 waves/block; wave w owns n-tiles w*4 .. w*4+3 (16 cols each).
// ---------------------------------------------------------------------------
__global__ __launch_bounds__(256) void k_uhl_pass(const float* __restrict__ x7,
                                                  const float* __restrict__ w7,
                                                  const float* __restrict__ cvec,
                                                  float* __restrict__ outp,
                                                  int mode) {
    __shared__ float uhl[16 * NCOL];    // 32 KiB strip
    __shared__ float fac[16 * 8];

    int t    = threadIdx.x;
    int lane = t & 31;
    int wv   = t >> 5;
    int r0   = blockIdx.x * 16;
    int mrow = r0 + (lane & 15);
    int khalf = (lane >> 4) << 1;       // lanes 0-15 -> K+0/K+1, lanes 16-31 -> K+2/K+3

    // Preload all 16 A fragments (16x4 fp32 per step) for this wave's rows.
    v2f a[16];
#pragma unroll
    for (int ks = 0; ks < 16; ++ks) {
        const float* ap = x7 + (size_t)mrow * 64 + ks * 4 + khalf;
        a[ks].x = ap[0];
        a[ks].y = ap[1];
    }

    for (int tt = 0; tt < 4; ++tt) {
        int n = (wv * 4 + tt) * 16 + (lane & 15);         // global column 0..511
        int j = n >> 6, e = n & 63;
        const float* bp = w7 + (size_t)j * 4096 + e;      // W7[j,d,e] column, stride_d=64
        V8 acc;
#pragma unroll
        for (int q = 0; q < 8; ++q) acc.f[q] = 0.f;
#pragma unroll
        for (int ks = 0; ks < 16; ++ks) {
            int k = ks * 4 + khalf;
            v2f b;
            b.x = bp[k * 64];
            b.y = bp[(k + 1) * 64];
            acc.v = __builtin_amdgcn_wmma_f32_16x16x4_f32(
                false, a[ks], false, b, (short)0, acc.v, false, false);
        }
        int mb = (lane >> 4) << 3;      // C/D: lanes 16-31 hold M+8
#pragma unroll
        for (int q = 0; q < 8; ++q)
            uhl[(mb + q) * NCOL + n] = acc.f[q];
    }
    __syncthreads();

    // squash factor per (row, j):  s = c_j * uhl_row;  g = c_j * s2/((1+s2)*sqrt(s2+eps))
    if (t < 128) {
        int rl = t >> 3, j = t & 7;
        const float* row = uhl + rl * NCOL + j * 64;
        float q = 0.f;
        for (int e = 0; e < 64; ++e) { float u = row[e]; q += u * u; }
        float c  = cvec[j];
        float s2 = c * c * q;
        float factor = s2 / ((1.f + s2) * sqrtf(s2 + CAP_EPS));
        fac[rl * 8 + j] = factor * c;
    }
    __syncthreads();

    if (mode == 0) {
        // per-block partial vsum[j][e] (deterministic, no atomics)
        int j = t >> 5, e0 = (t & 31) * 2;
        float a0 = 0.f, a1 = 0.f;
        const float* col = uhl + j * 64 + e0;
        for (int rl = 0; rl < 16; ++rl) {
            float g = fac[rl * 8 + j];
            a0 += g * col[rl * NCOL];
            a1 += g * col[rl * NCOL + 1];
        }
        float* pp = outp + (size_t)blockIdx.x * NCOL + j * 64 + e0;
        pp[0] = a0; pp[1] = a1;
    } else {
        // final output: r = bb*512+s (natural), jb = s*32+bb; out[(j, jb, e)]
        for (int idx = t; idx < 16 * NCOL; idx += 256) {
            int rl = idx >> 9;
            int n  = idx & 511;
            int j = n >> 6, e = n & 63;
            int r  = r0 + rl;
            int jb = (r & 511) * 32 + (r >> 9);
            outp[(size_t)j * (JB * 64) + (size_t)jb * 64 + e] =
                fac[rl * 8 + j] * uhl[rl * NCOL + n];
        }
    }
}

// ---------------------------------------------------------------------------
// Reduce 1024 vsum partials -> 32 partials (stage 1)
// ---------------------------------------------------------------------------
__global__ __launch_bounds__(256) void k_vred1(const float* __restrict__ vpart,
                                               float* __restrict__ vpart2) {
    int b = blockIdx.x;                 // 0..31
    int t = threadIdx.x;                // 0..255
    for (int v = t; v < NCOL; v += 256) {
        const float* p = vpart + (size_t)b * 32 * NCOL + v;
        float s = 0.f;
        for (int q = 0; q < 32; ++q) s += p[q * NCOL];
        vpart2[b * NCOL + v] = s;
    }
}

// ---------------------------------------------------------------------------
// Reduce 32 partials -> vmean, then B[i,j] += sum_e uhm[i,j,e]*vmean[j,e]
// ---------------------------------------------------------------------------
__global__ __launch_bounds__(512) void k_vred2(const float* __restrict__ vpart2,
                                               const float* __restrict__ uhm,
                                               float* __restrict__ Bmat) {
    __shared__ float vmean[NCOL];
    int t = threadIdx.x;                // 0..511 -> (j,e)
    float s = 0.f;
    for (int b = 0; b < 32; ++b) s += vpart2[b * NCOL + t];
    vmean[t] = s * (1.0f / (float)JB);
    __syncthreads();
    if (t < 64) {
        int j = t & 7;
        const float* up = uhm + t * 64;
        const float* vp = vmean + j * 64;
        float d = 0.f;
        for (int e = 0; e < 64; ++e) d += up[e] * vp[e];
        Bmat[t] += d;
    }
}

// ---------------------------------------------------------------------------
extern "C" void kernel_launch(void* const* d_in, const int* in_sizes, int n_in,
                              void* d_out, int out_size, void* d_ws, size_t ws_size,
                              hipStream_t stream) {
    const float* x = (const float*)d_in[0];   // (8, 32, 512, 64)
    const float* W = (const float*)d_in[1];   // (8, 8, 64, 64)
    float* out = (float*)d_out;
    float* ws  = (float*)d_ws;

    // workspace layout (floats): ~2.2 MiB total
    float* vpart  = ws;                       // 1024*512
    float* xpart  = vpart + STRIPS * NCOL;    // 8*16*64 = 8192
    float* vpart2 = xpart + 8192;             // 32*512  = 16384
    float* uhm    = vpart2 + 16384;           // 8*8*64  = 4096
    float* Bmat   = uhm + 4096;               // 64
    float* cvec   = Bmat + 64;                // 8

    const float* x7 = x + (size_t)7 * JB * IN_DIM;          // u_last rows (natural order)
    const float* w7 = W + (size_t)7 * NUM_OUT * IN_DIM * OUT_DIM;

    k_xpart<<<dim3(16, 8), 256, 0, stream>>>(x, xpart);
    k_uhm<<<1, 256, 0, stream>>>(xpart, W, uhm, Bmat);

    // routing iterations 1 and 2 update B; iteration 3's softmax feeds the output
    for (int it = 0; it < 2; ++it) {
        k_softmax<<<1, 64, 0, stream>>>(Bmat, cvec);
        k_uhl_pass<<<STRIPS, 256, 0, stream>>>(x7, w7, cvec, vpart, 0);
        k_vred1<<<32, 256, 0, stream>>>(vpart, vpart2);
        k_vred2<<<1, 512, 0, stream>>>(vpart2, uhm, Bmat);
    }
    k_softmax<<<1, 64, 0, stream>>>(Bmat, cvec);
    k_uhl_pass<<<STRIPS, 256, 0, stream>>>(x7, w7, cvec, out, 1);
}